// AdvancedGraphormer_33337536151688
// MI455X (gfx1250) — compile-verified
//
#include <hip/hip_runtime.h>
#include <math.h>

// =====================================================================
// AdvancedGraphormer forward on gfx1250 (MI455X).
// All dense linears run through a generic 4-segment gather/concat GEMM
// built on V_WMMA_F32_16X16X4_F32 (full f32 to match reference numerics).
// Each wave produces a 16x64 output strip (4 independent accumulators)
// with pure pointer-increment addressing in the K loop — no predication,
// no 64-bit multiplies between WMMAs.
// Global attention is a fused per-(graph,head) LDS kernel with 2 WMMA
// stages. Segment softmax uses ordered-uint atomicMax + float atomicAdd.
//
// Param leaves are assumed pytree-flattened (sorted dict keys, JAX style)
// after the six positional inputs.
// =====================================================================

typedef float v2f __attribute__((ext_vector_type(2)));
typedef float v8f __attribute__((ext_vector_type(8)));

constexpr int kN   = 8192;     // nodes
constexpr int kE   = 131072;   // edges
constexpr int kHid = 256;
constexpr int kDe  = 128;
constexpr int kB   = 128;      // graphs
constexpr int kNp  = 64;       // nodes per graph
constexpr int kEp  = 1024;     // edges per graph
constexpr float kBnScale = 0.9999950000374997f;   // 1/sqrt(1+1e-5)
constexpr float kRs32    = 0.17677669529663687f;  // 1/sqrt(32)

__device__ __forceinline__ float wsum(float v) {
#pragma unroll
  for (int m = 16; m > 0; m >>= 1) v += __shfl_xor(v, m, 32);
  return v;
}
__device__ __forceinline__ unsigned f2ord(float f) {
  unsigned u = __float_as_uint(f);
  return (u & 0x80000000u) ? ~u : (u ^ 0x80000000u);
}
__device__ __forceinline__ float ord2f(unsigned e) {
  const unsigned u = (e & 0x80000000u) ? (e ^ 0x80000000u) : ~e;
  return __uint_as_float(u);
}
__device__ __forceinline__ float act_apply(float v, int act) {
  if (act == 1)      v = fmaxf(v, 0.f);
  else if (act == 2) v = fmaxf(v, 0.f) * kBnScale;
  else if (act == 3) v = 1.f / (1.f + expf(-v));
  else if (act == 4) v = 0.5f * v * (1.f + erff(v * 0.70710678118654752f));
  return v;
}

// ------------------------------------------------------------------
// Generic WMMA GEMM:  out = act(alpha * concat(segs) @ W + bias) + add0 + add1
// Up to 4 A-segments: {ptr, optional row-index array, ld, kwidth}.
// REQUIREMENTS (all call sites satisfy): M % 16 == 0, N % 64 == 0,
// each kw % 4 == 0, rows 8-byte aligned.
// One wave computes a 16x64 strip: 4 N-tiles sharing one A fragment,
// 4 independent accumulator chains. K stepped by 4 per WMMA.
// act: 0 none, 1 relu, 2 relu*bnscale, 3 sigmoid, 4 exact gelu.
// ------------------------------------------------------------------
__global__ __launch_bounds__(128) void gemm_wmma_f32(
    const float* a0, const int* r0, int ld0, int kw0,
    const float* a1, const int* r1, int ld1, int kw1,
    const float* a2, const int* r2, int ld2, int kw2,
    const float* a3, const int* r3, int ld3, int kw3,
    const float* __restrict__ w, const float* __restrict__ bias,
    const float* __restrict__ add0, const float* __restrict__ add1,
    float* __restrict__ out, int M, int Nc, float alpha, int act)
{
  const int lane  = threadIdx.x & 31;
  const int wave  = threadIdx.x >> 5;
  const int r     = lane & 15;
  const int half  = lane >> 4;
  const int quadsN = Nc >> 6;                 // N multiple of 64
  const int total  = (M >> 4) * quadsN;
  const int t = blockIdx.x * 4 + wave;        // wave-uniform
  if (t >= total) return;                     // whole wave exits together
  const int tm = t / quadsN;
  const int tq = t - tm * quadsN;
  const int row  = tm * 16 + r;
  const int col0 = tq * 64 + r;               // sub-tile 0 column for this lane

  v8f acc0 = {0.f,0.f,0.f,0.f,0.f,0.f,0.f,0.f};
  v8f acc1 = acc0, acc2 = acc0, acc3 = acc0;
  int kg = 0;
#define GEMM_SEG(AP, RI, LD, KW)                                              \
  if (KW) {                                                                   \
    const float* ra = (AP) + (size_t)((RI) ? (RI)[row] : row) * (LD)          \
                      + 2 * half;                                             \
    const float* wq = w + (size_t)(kg + 2 * half) * Nc + col0;                \
    const size_t stepW = (size_t)4 * Nc;                                      \
    for (int kin = 0; kin < (KW); kin += 4) {                                 \
      const v2f a = *(const v2f*)ra;  ra += 4;                                \
      v2f b0, b1, b2, b3;                                                     \
      b0.x = wq[0];       b0.y = wq[Nc];                                      \
      b1.x = wq[16];      b1.y = wq[16 + Nc];                                 \
      b2.x = wq[32];      b2.y = wq[32 + Nc];                                 \
      b3.x = wq[48];      b3.y = wq[48 + Nc];                                 \
      wq += stepW;                                                            \
      acc0 = __builtin_amdgcn_wmma_f32_16x16x4_f32(false, a, false, b0,       \
                                                   (short)0, acc0, false, false); \
      acc1 = __builtin_amdgcn_wmma_f32_16x16x4_f32(false, a, false, b1,       \
                                                   (short)0, acc1, false, false); \
      acc2 = __builtin_amdgcn_wmma_f32_16x16x4_f32(false, a, false, b2,       \
                                                   (short)0, acc2, false, false); \
      acc3 = __builtin_amdgcn_wmma_f32_16x16x4_f32(false, a, false, b3,       \
                                                   (short)0, acc3, false, false); \
    }                                                                         \
    kg += (KW);                                                               \
  }
  GEMM_SEG(a0, r0, ld0, kw0)
  GEMM_SEG(a1, r1, ld1, kw1)
  GEMM_SEG(a2, r2, ld2, kw2)
  GEMM_SEG(a3, r3, ld3, kw3)
#undef GEMM_SEG

#define GEMM_STORE(ACC, SUB)                                                  \
  {                                                                           \
    const int col = col0 + (SUB) * 16;                                        \
    const float bv = bias ? bias[col] : 0.f;                                  \
    _Pragma("unroll")                                                         \
    for (int i = 0; i < 8; ++i) {                                             \
      float v = act_apply(ACC[i] * alpha + bv, act);                          \
      const size_t idx = (size_t)(tm * 16 + half * 8 + i) * Nc + col;         \
      if (add0) v += add0[idx];                                               \
      if (add1) v += add1[idx];                                               \
      out[idx] = v;                                                           \
    }                                                                         \
  }
  GEMM_STORE(acc0, 0)
  GEMM_STORE(acc1, 1)
  GEMM_STORE(acc2, 2)
  GEMM_STORE(acc3, 3)
#undef GEMM_STORE
}

// ------------------------------------------------------------------
// Fused dense global attention: one block per (graph b, head h).
// qs scaled by 1/sqrt(dh) at load. scores and P kept in LDS; two WMMA
// stages (Q*K^T then P*V).
// ------------------------------------------------------------------
__global__ __launch_bounds__(256) void global_attn_kernel(
    const float* __restrict__ Q, const float* __restrict__ K,
    const float* __restrict__ V, const float* __restrict__ Bb,
    float* __restrict__ AO)
{
  __shared__ float qs[64][36];
  __shared__ float ks[64][36];
  __shared__ float vs[64][36];
  __shared__ float ss[64][68];
  const int bh = blockIdx.x;
  const int b = bh >> 3, h = bh & 7;
  const int tid = threadIdx.x;
  for (int idx = tid; idx < 64 * 32; idx += 256) {
    const int i = idx >> 5, d = idx & 31;
    const size_t g = (size_t)(b * 64 + i) * kHid + h * 32 + d;
    qs[i][d] = Q[g] * kRs32;
    ks[i][d] = K[g];
    vs[i][d] = V[g];
  }
  __syncthreads();
  const int wave = tid >> 5, lane = tid & 31, r = lane & 15, half = lane >> 4;

  // S = (Q/sqrt(dh)) * K^T + bias     (16 tiles of 16x16, K=32)
  for (int tt = wave; tt < 16; tt += 8) {
    const int tm = tt >> 2, tn = tt & 3;
    v8f acc = {0.f,0.f,0.f,0.f,0.f,0.f,0.f,0.f};
    for (int kk = 0; kk < 32; kk += 4) {
      const int k0 = kk + 2 * half;
      v2f a; a.x = qs[tm * 16 + r][k0]; a.y = qs[tm * 16 + r][k0 + 1];
      v2f bb; bb.x = ks[tn * 16 + r][k0]; bb.y = ks[tn * 16 + r][k0 + 1];
      acc = __builtin_amdgcn_wmma_f32_16x16x4_f32(false, a, false, bb,
                                                  (short)0, acc, false, false);
    }
    const float* bp = Bb + (size_t)(b * 8 + h) * 64 * 64;
#pragma unroll
    for (int i = 0; i < 8; ++i) {
      const int ri = tm * 16 + half * 8 + i, ci = tn * 16 + r;
      ss[ri][ci] = acc[i] + bp[ri * 64 + ci];
    }
  }
  __syncthreads();
  if (tid < 64) {
    float mx = -3.402823466e38f;
    for (int j = 0; j < 64; ++j) mx = fmaxf(mx, ss[tid][j]);
    float sum = 0.f;
    for (int j = 0; j < 64; ++j) { float e = expf(ss[tid][j] - mx); ss[tid][j] = e; sum += e; }
    const float inv = 1.f / sum;
    for (int j = 0; j < 64; ++j) ss[tid][j] *= inv;
  }
  __syncthreads();
  // O = P * V   (8 tiles, K=64)
  {
    const int tm = wave >> 1, tn = wave & 1;
    v8f acc = {0.f,0.f,0.f,0.f,0.f,0.f,0.f,0.f};
    for (int kk = 0; kk < 64; kk += 4) {
      const int k0 = kk + 2 * half;
      v2f a; a.x = ss[tm * 16 + r][k0]; a.y = ss[tm * 16 + r][k0 + 1];
      v2f bb; bb.x = vs[k0][tn * 16 + r]; bb.y = vs[k0 + 1][tn * 16 + r];
      acc = __builtin_amdgcn_wmma_f32_16x16x4_f32(false, a, false, bb,
                                                  (short)0, acc, false, false);
    }
#pragma unroll
    for (int i = 0; i < 8; ++i) {
      const int ri = tm * 16 + half * 8 + i, ci = tn * 16 + r;
      AO[(size_t)(b * 64 + ri) * kHid + h * 32 + ci] = acc[i];
    }
  }
}

// ---------------- small helper kernels ----------------
__global__ void deg_kernel(const int* __restrict__ src, int* deg, int n) {
  int t = blockIdx.x * blockDim.x + threadIdx.x;
  if (t < n) atomicAdd(&deg[src[t]], 1);
}
__global__ void spdc_kernel(const int* __restrict__ spd, int* out, int n) {
  int t = blockIdx.x * blockDim.x + threadIdx.x;
  if (t < n) { int v = spd[t]; out[t] = v < 0 ? 0 : (v > 31 ? 31 : v); }
}
__global__ void xin_kernel(const float* __restrict__ xc, const float* __restrict__ demb,
                           const int* __restrict__ deg, float* __restrict__ out, int n) {
  int t = blockIdx.x * blockDim.x + threadIdx.x;
  if (t >= n) return;
  const int i = t >> 8, c = t & 255;
  int d = deg[i]; d = d > 63 ? 63 : d;
  out[t] = xc[t] + demb[d * kHid + c];
}
__global__ void accum_kernel(float* __restrict__ dst, const float* __restrict__ s, int n) {
  int t = blockIdx.x * blockDim.x + threadIdx.x;
  if (t < n) dst[t] += s[t];
}
__global__ __launch_bounds__(256) void rowln_kernel(const float* __restrict__ in,
                                                    float* __restrict__ out, int rows, int D) {
  const int row = blockIdx.x * 8 + (threadIdx.x >> 5);
  if (row >= rows) return;
  const int lane = threadIdx.x & 31;
  const float* p = in + (size_t)row * D;
  float s = 0.f, s2 = 0.f;
  for (int c = lane; c < D; c += 32) { float v = p[c]; s += v; s2 += v * v; }
  s = wsum(s); s2 = wsum(s2);
  const float mu = s / D;
  const float var = s2 / D - mu * mu;
  const float inv = 1.f / sqrtf(var + 1e-5f);
  float* o = out + (size_t)row * D;
  for (int c = lane; c < D; c += 32) o[c] = (p[c] - mu) * inv;
}
// ea = LN(gate*ea_hc + (1-gate)*spd_emb[spd])   (D=128, wave per row)
__global__ __launch_bounds__(256) void fusion_ln_kernel(
    const float* __restrict__ gate, const float* __restrict__ eahc,
    const float* __restrict__ semb, const int* __restrict__ spdc,
    float* __restrict__ ea, int nE) {
  const int e = blockIdx.x * 8 + (threadIdx.x >> 5);
  if (e >= nE) return;
  const int lane = threadIdx.x & 31;
  const float* sp = semb + (size_t)spdc[e] * kDe;
  float vals[4]; float s = 0.f, s2 = 0.f;
#pragma unroll
  for (int j = 0; j < 4; ++j) {
    const int c = lane + j * 32;
    const float g = gate[(size_t)e * kDe + c];
    const float v = g * eahc[(size_t)e * kDe + c] + (1.f - g) * sp[c];
    vals[j] = v; s += v; s2 += v * v;
  }
  s = wsum(s); s2 = wsum(s2);
  const float mu = s * (1.f / kDe);
  const float inv = 1.f / sqrtf(s2 * (1.f / kDe) - mu * mu + 1e-5f);
#pragma unroll
  for (int j = 0; j < 4; ++j) ea[(size_t)e * kDe + lane + j * 32] = (vals[j] - mu) * inv;
}
// ea = LN(ea + sigmoid(raw[:,128:]) * raw[:,:128])
__global__ __launch_bounds__(256) void euupd_ln_kernel(
    const float* __restrict__ raw, float* __restrict__ ea, int nE) {
  const int e = blockIdx.x * 8 + (threadIdx.x >> 5);
  if (e >= nE) return;
  const int lane = threadIdx.x & 31;
  float vals[4]; float s = 0.f, s2 = 0.f;
#pragma unroll
  for (int j = 0; j < 4; ++j) {
    const int c = lane + j * 32;
    const float dlt = raw[(size_t)e * kHid + c];
    const float gl  = raw[(size_t)e * kHid + 128 + c];
    const float v = ea[(size_t)e * kDe + c] + dlt / (1.f + expf(-gl));
    vals[j] = v; s += v; s2 += v * v;
  }
  s = wsum(s); s2 = wsum(s2);
  const float mu = s * (1.f / kDe);
  const float inv = 1.f / sqrtf(s2 * (1.f / kDe) - mu * mu + 1e-5f);
#pragma unroll
  for (int j = 0; j < 4; ++j) ea[(size_t)e * kDe + lane + j * 32] = (vals[j] - mu) * inv;
}
// local attention: scores + segment max
__global__ void lscore_kernel(const float* __restrict__ q, const float* __restrict__ k,
                              const float* __restrict__ e, const int* __restrict__ src,
                              const int* __restrict__ dst, float* __restrict__ s,
                              unsigned* __restrict__ m) {
  int t = blockIdx.x * blockDim.x + threadIdx.x;
  if (t >= kE * 8) return;
  const int ed = t >> 3, h = t & 7;
  const float* qp = q + (size_t)dst[ed] * kHid + h * 32;
  const float* kp = k + (size_t)src[ed] * kHid + h * 32;
  const float* ep = e + (size_t)ed * kHid + h * 32;
  float acc = 0.f;
#pragma unroll
  for (int d = 0; d < 32; ++d) acc += qp[d] * (kp[d] + ep[d]);
  acc *= kRs32;
  s[t] = acc;
  atomicMax(&m[(size_t)dst[ed] * 8 + h], f2ord(acc));
}
__global__ void lexp_kernel(float* __restrict__ s, const unsigned* __restrict__ m,
                            float* __restrict__ z, const int* __restrict__ dst) {
  int t = blockIdx.x * blockDim.x + threadIdx.x;
  if (t >= kE * 8) return;
  const int ed = t >> 3, h = t & 7;
  const float ex = expf(s[t] - ord2f(m[(size_t)dst[ed] * 8 + h]));
  s[t] = ex;
  atomicAdd(&z[(size_t)dst[ed] * 8 + h], ex);
}
__global__ void lagg_kernel(const float* __restrict__ s, const float* __restrict__ z,
                            const float* __restrict__ v, const float* __restrict__ e,
                            const int* __restrict__ src, const int* __restrict__ dst,
                            float* __restrict__ agg) {
  int t = blockIdx.x * blockDim.x + threadIdx.x;
  if (t >= kE * kHid) return;
  const int ed = t >> 8, c = t & 255, h = c >> 5;
  const int dn = dst[ed];
  const float a = s[(size_t)ed * 8 + h] / (z[(size_t)dn * 8 + h] + 1e-16f);
  const float val = a * (v[(size_t)src[ed] * kHid + c] + e[(size_t)ed * kHid + c]);
  atomicAdd(&agg[(size_t)dn * kHid + c], val);
}
// node-edge cross attention (seg = src, 4 heads of 32)
__global__ void xscore_kernel(const float* __restrict__ qx, const float* __restrict__ kx,
                              const int* __restrict__ src, float* __restrict__ s,
                              unsigned* __restrict__ m) {
  int t = blockIdx.x * blockDim.x + threadIdx.x;
  if (t >= kE * 4) return;
  const int ed = t >> 2, h = t & 3;
  const float* qp = qx + (size_t)src[ed] * kDe + h * 32;
  const float* kp = kx + (size_t)ed * kDe + h * 32;
  float acc = 0.f;
#pragma unroll
  for (int d = 0; d < 32; ++d) acc += qp[d] * kp[d];
  acc *= kRs32;
  s[t] = acc;
  atomicMax(&m[(size_t)src[ed] * 4 + h], f2ord(acc));
}
__global__ void xexp_kernel(float* __restrict__ s, const unsigned* __restrict__ m,
                            float* __restrict__ z, const int* __restrict__ src) {
  int t = blockIdx.x * blockDim.x + threadIdx.x;
  if (t >= kE * 4) return;
  const int ed = t >> 2, h = t & 3;
  const float ex = expf(s[t] - ord2f(m[(size_t)src[ed] * 4 + h]));
  s[t] = ex;
  atomicAdd(&z[(size_t)src[ed] * 4 + h], ex);
}
__global__ void xagg_kernel(const float* __restrict__ s, const float* __restrict__ z,
                            const float* __restrict__ vx, const int* __restrict__ src,
                            float* __restrict__ agg) {
  int t = blockIdx.x * blockDim.x + threadIdx.x;
  if (t >= kE * kDe) return;
  const int ed = t >> 7, c = t & 127, h = c >> 5;
  const int sn = src[ed];
  const float wgt = s[(size_t)ed * 4 + h] / (z[(size_t)sn * 4 + h] + 1e-16f);
  atomicAdd(&agg[(size_t)sn * kDe + c], wgt * vx[(size_t)ed * kDe + c]);
}
__global__ void adj_kernel(const int* __restrict__ src, const int* __restrict__ dst,
                           const int* __restrict__ spd, int* __restrict__ adj, int n) {
  int t = blockIdx.x * blockDim.x + threadIdx.x;
  if (t >= n) return;
  const int s = src[t], d = dst[t];
  atomicAdd(&adj[(size_t)(s >> 6) * 4096 + (s & 63) * 64 + (d & 63)], spd[t]);
}
// bias[b,h,i,j] = dist_bias[clip(adj)][h] + rbf(d3) @ rbf_w + rbf_b
__global__ void bias_kernel(const float* __restrict__ pos, const int* __restrict__ adj,
                            const float* __restrict__ dist_bias,
                            const float* __restrict__ rbf_w, const float* __restrict__ rbf_b,
                            float* __restrict__ bias) {
  int t = blockIdx.x * blockDim.x + threadIdx.x;
  if (t >= kB * 64 * 64) return;
  const int j = t & 63, i = (t >> 6) & 63, b = t >> 12;
  const float* pi = pos + (size_t)(b * 64 + i) * 3;
  const float* pj = pos + (size_t)(b * 64 + j) * 3;
  const float dx = pi[0] - pj[0], dy = pi[1] - pj[1], dz = pi[2] - pj[2];
  const float d = sqrtf(dx * dx + dy * dy + dz * dz + 1e-12f);
  int a = adj[t]; a = a < 0 ? 0 : (a > 31 ? 31 : a);
  float acc[8];
#pragma unroll
  for (int h = 0; h < 8; ++h) acc[h] = dist_bias[a * 8 + h] + rbf_b[h];
  const float step = 20.f / 31.f;
  const float coeff = -0.5f / (step * step);
  for (int rr = 0; rr < 32; ++rr) {
    const float o = step * rr;
    const float e = expf(coeff * (d - o) * (d - o));
#pragma unroll
    for (int h = 0; h < 8; ++h) acc[h] += e * rbf_w[rr * 8 + h];
  }
#pragma unroll
  for (int h = 0; h < 8; ++h)
    bias[((size_t)(b * 8 + h) * 64 + i) * 64 + j] = acc[h];
}
__global__ void pool_nodes_kernel(const float* __restrict__ x, float* __restrict__ gin) {
  int t = blockIdx.x * blockDim.x + threadIdx.x;
  if (t >= kB * kHid) return;
  const int b = t >> 8, c = t & 255;
  float s = 0.f, mx = -3.402823466e38f;
  for (int i = 0; i < kNp; ++i) {
    const float v = x[(size_t)(b * kNp + i) * kHid + c];
    s += v; mx = fmaxf(mx, v);
  }
  gin[(size_t)b * 512 + c] = s * (1.f / kNp);
  gin[(size_t)b * 512 + 256 + c] = mx;
}
__global__ void pool_mean_kernel(const float* __restrict__ src, float* __restrict__ out,
                                 int rows, int D, int Bc) {
  int t = blockIdx.x * blockDim.x + threadIdx.x;
  if (t >= Bc * D) return;
  const int b = t / D, c = t - b * D;
  float s = 0.f;
  for (int i = 0; i < rows; ++i) s += src[(size_t)(b * rows + i) * D + c];
  out[t] = s / rows;
}
__global__ void head3_kernel(const float* __restrict__ h2, const float* __restrict__ w,
                             const float* __restrict__ b, float* __restrict__ out) {
  int t = blockIdx.x * blockDim.x + threadIdx.x;
  if (t >= kB) return;
  float s = b[0];
  for (int c = 0; c < kDe; ++c) s += h2[(size_t)t * kDe + c] * w[c];
  out[t] = s;
}

// =====================================================================
static inline int ceil_div(int a, int b) { return (a + b - 1) / b; }

extern "C" void kernel_launch(void* const* d_in, const int* in_sizes, int n_in,
                              void* d_out, int out_size, void* d_ws, size_t ws_size,
                              hipStream_t stream) {
  (void)in_sizes; (void)n_in; (void)out_size; (void)ws_size;
  const float* X   = (const float*)d_in[0];
  const float* POS = (const float*)d_in[1];
  const float* EAt = (const float*)d_in[2];
  const int*   EI  = (const int*)d_in[3];
  const int*   SPD = (const int*)d_in[4];
  const int* src = EI;
  const int* dst = EI + kE;
#define P(i) ((const float*)d_in[6 + (i)])
  // sorted-key leaf order:
  // 0/1 edge_proj.b/w  2/3 fusion_gate.b/w  4/5 gep.b/w
  // 6 glb.dist_bias 7/8 glb.ffn1.b/w 9/10 glb.ffn2.b/w 11/12 glb.k.b/w
  // 13/14 glb.o.b/w 15/16 glb.q.b/w 17/18 glb.rbf.b/w 19/20 glb.v.b/w
  // 21/22 gpp.b/w 23/24 head1.b/w 25/26 head2.b/w 27/28 head3.b/w
  // 29+16l: deg_emb,e_w,eu.b,eu.w,ffn1.b,ffn1.w,ffn2.b,ffn2.w,k.b,k.w,
  //         q.b,q.w,skip.b,skip.w,v.b,v.w
  // 93/94 node_proj.b/w 95/96 nrp.b/w 97 spatial_emb
  // 98..105 xattn k.b,k.w,o.b,o.w,q.b,q.w,v.b,v.w

  char* base = (char*)d_ws;
  size_t off = 0;
  auto alloc = [&](size_t bytes) -> void* {
    void* p = base + off; off += (bytes + 255) & ~(size_t)255; return p;
  };
  int*   deg    = (int*)alloc((size_t)kN * 4);
  int*   spdc   = (int*)alloc((size_t)kE * 4);
  int*   adj    = (int*)alloc((size_t)kB * 4096 * 4);
  float* h0     = (float*)alloc((size_t)kN * kHid * 4);
  float* eahc   = (float*)alloc((size_t)kE * kDe * 4);
  float* gate   = (float*)alloc((size_t)kE * kDe * 4);
  float* ea     = (float*)alloc((size_t)kE * kDe * 4);
  float* eacomb = (float*)alloc((size_t)kE * kDe * 4);
  float* xc     = (float*)alloc((size_t)kN * kHid * 4);
  float* xcomb  = (float*)alloc((size_t)kN * kHid * 4);
  float* xin    = (float*)alloc((size_t)kN * kHid * 4);
  float* qb     = (float*)alloc((size_t)kN * kHid * 4);
  float* kb     = (float*)alloc((size_t)kN * kHid * 4);
  float* vb     = (float*)alloc((size_t)kN * kHid * 4);
  float* ebig   = (float*)alloc((size_t)kE * kHid * 4);   // e-proj, then eu raw
  float* sc     = (float*)alloc((size_t)kE * 8 * 4);
  unsigned* ms  = (unsigned*)alloc((size_t)kN * 8 * 4);
  float* zs     = (float*)alloc((size_t)kN * 8 * 4);
  float* agg    = (float*)alloc((size_t)kN * kHid * 4);
  float* xat    = (float*)alloc((size_t)kN * kHid * 4);
  float* xnb    = (float*)alloc((size_t)kN * kHid * 4);
  float* ffn    = (float*)alloc((size_t)kN * 1024 * 4);
  float* biasb  = (float*)alloc((size_t)kB * 8 * 64 * 64 * 4);
  float* hgb    = (float*)alloc((size_t)kN * kHid * 4);
  float* hg2    = (float*)alloc((size_t)kN * kHid * 4);
  float* tln    = (float*)alloc((size_t)kN * kHid * 4);
  float* aob    = (float*)alloc((size_t)kN * kHid * 4);
  float* hnp    = (float*)alloc((size_t)kN * kDe * 4);
  float* qx     = (float*)alloc((size_t)kN * kDe * 4);
  float* s2     = (float*)alloc((size_t)kE * 4 * 4);
  unsigned* m2  = (unsigned*)alloc((size_t)kN * 4 * 4);
  float* z2     = (float*)alloc((size_t)kN * 4 * 4);
  float* agg2   = (float*)alloc((size_t)kN * kDe * 4);
  float* smart  = (float*)alloc((size_t)kN * kDe * 4);
  float* gin    = (float*)alloc((size_t)kB * 512 * 4);
  float* gpool  = (float*)alloc((size_t)kB * kDe * 4);
  float* emean  = (float*)alloc((size_t)kB * kDe * 4);
  float* gedge  = (float*)alloc((size_t)kB * kDe * 4);
  float* npool  = (float*)alloc((size_t)kB * kDe * 4);
  float* spool  = (float*)alloc((size_t)kB * kDe * 4);
  float* h1     = (float*)alloc((size_t)kB * 256 * 4);
  float* h2b    = (float*)alloc((size_t)kB * kDe * 4);
  float* kx = eahc;   // reuse after fusion stage
  float* vx = gate;   // reuse after fusion stage

  auto gemm = [&](const float* a0, const int* r0, int ld0, int kw0,
                  const float* a1, const int* r1, int ld1, int kw1,
                  const float* a2, const int* r2, int ld2, int kw2,
                  const float* a3, const int* r3, int ld3, int kw3,
                  const float* w, const float* bias, const float* add0, const float* add1,
                  float* out, int M, int Nc, float alpha, int act) {
    const int tiles = (M / 16) * (Nc / 64);   // N % 64 == 0 at all call sites
    gemm_wmma_f32<<<ceil_div(tiles, 4), 128, 0, stream>>>(
        a0, r0, ld0, kw0, a1, r1, ld1, kw1, a2, r2, ld2, kw2, a3, r3, ld3, kw3,
        w, bias, add0, add1, out, M, Nc, alpha, act);
  };
  auto gemm1 = [&](const float* a, int K, const float* w, const float* bias,
                   const float* add0, const float* add1, float* out,
                   int M, int Nc, float alpha, int act) {
    gemm(a, nullptr, K, K, nullptr, nullptr, 0, 0, nullptr, nullptr, 0, 0,
         nullptr, nullptr, 0, 0, w, bias, add0, add1, out, M, Nc, alpha, act);
  };

  // ---- preamble ----
  hipMemsetAsync(deg, 0, (size_t)kN * 4, stream);
  deg_kernel<<<ceil_div(kE, 256), 256, 0, stream>>>(src, deg, kE);
  spdc_kernel<<<ceil_div(kE, 256), 256, 0, stream>>>(SPD, spdc, kE);

  gemm1(X, 64, P(94), P(93), nullptr, nullptr, h0, kN, kHid, 1.f, 2);      // node_proj
  gemm1(EAt, 16, P(1), P(0), nullptr, nullptr, eahc, kE, kDe, 1.f, 2);     // edge_proj
  gemm(eahc, nullptr, kDe, kDe, P(97), spdc, kDe, kDe,                     // fusion gate
       nullptr, nullptr, 0, 0, nullptr, nullptr, 0, 0,
       P(3), P(2), nullptr, nullptr, gate, kE, kDe, 1.f, 3);
  fusion_ln_kernel<<<ceil_div(kE, 8), 256, 0, stream>>>(gate, eahc, P(97), spdc, ea, kE);

  hipMemsetAsync(xcomb, 0, (size_t)kN * kHid * 4, stream);
  hipMemsetAsync(eacomb, 0, (size_t)kE * kDe * 4, stream);

  // ---- 4 transformer-conv layers ----
  const float* curx = h0;
  for (int l = 0; l < 4; ++l) {
    const int L = 29 + 16 * l;
    xin_kernel<<<ceil_div(kN * kHid, 256), 256, 0, stream>>>(curx, P(L + 0), deg, xin, kN * kHid);
    gemm1(xin, kHid, P(L + 11), P(L + 10), nullptr, nullptr, qb, kN, kHid, kBnScale, 0);
    gemm1(xin, kHid, P(L + 9),  P(L + 8),  nullptr, nullptr, kb, kN, kHid, kBnScale, 0);
    gemm1(xin, kHid, P(L + 15), P(L + 14), nullptr, nullptr, vb, kN, kHid, kBnScale, 0);
    gemm1(ea, kDe, P(L + 1), nullptr, nullptr, nullptr, ebig, kE, kHid, 1.f, 0); // e = ea@e_w
    hipMemsetAsync(ms, 0, (size_t)kN * 8 * 4, stream);
    hipMemsetAsync(zs, 0, (size_t)kN * 8 * 4, stream);
    hipMemsetAsync(agg, 0, (size_t)kN * kHid * 4, stream);
    lscore_kernel<<<ceil_div(kE * 8, 256), 256, 0, stream>>>(qb, kb, ebig, src, dst, sc, ms);
    lexp_kernel<<<ceil_div(kE * 8, 256), 256, 0, stream>>>(sc, ms, zs, dst);
    lagg_kernel<<<ceil_div(kE * kHid, 256), 256, 0, stream>>>(sc, zs, vb, ebig, src, dst, agg);
    // x_attn = x_in + agg + skip(hn)
    gemm1(xin, kHid, P(L + 13), P(L + 12), xin, agg, xat, kN, kHid, kBnScale, 0);
    gemm1(xat, kHid, P(L + 5), P(L + 4), nullptr, nullptr, ffn, kN, 1024, kBnScale, 4);
    gemm1(ffn, 1024, P(L + 7), P(L + 6), xat, nullptr, xc, kN, kHid, 1.f, 0);
    rowln_kernel<<<ceil_div(kN, 8), 256, 0, stream>>>(xc, xnb, kN, kHid);
    // eu: relu(concat(xn[src], xn[dst], ea) @ W + b)
    gemm(xnb, src, kHid, kHid, xnb, dst, kHid, kHid, ea, nullptr, kDe, kDe,
         nullptr, nullptr, 0, 0, P(L + 3), P(L + 2), nullptr, nullptr,
         ebig, kE, kHid, 1.f, 1);
    euupd_ln_kernel<<<ceil_div(kE, 8), 256, 0, stream>>>(ebig, ea, kE);
    if (l >= 1) {
      accum_kernel<<<ceil_div(kN * kHid, 256), 256, 0, stream>>>(xcomb, xc, kN * kHid);
      accum_kernel<<<ceil_div(kE * kDe, 256), 256, 0, stream>>>(eacomb, ea, kE * kDe);
    }
    curx = xc;
  }

  // ---- global Graphormer block ----
  hipMemsetAsync(adj, 0, (size_t)kB * 4096 * 4, stream);
  adj_kernel<<<ceil_div(kE, 256), 256, 0, stream>>>(src, dst, SPD, adj, kE);
  bias_kernel<<<ceil_div(kB * 4096, 256), 256, 0, stream>>>(POS, adj, P(6), P(18), P(17), biasb);
  rowln_kernel<<<ceil_div(kN, 8), 256, 0, stream>>>(xcomb, hgb, kN, kHid);
  gemm1(hgb, kHid, P(16), P(15), nullptr, nullptr, qb, kN, kHid, 1.f, 0);
  gemm1(hgb, kHid, P(12), P(11), nullptr, nullptr, kb, kN, kHid, 1.f, 0);
  gemm1(hgb, kHid, P(20), P(19), nullptr, nullptr, vb, kN, kHid, 1.f, 0);
  global_attn_kernel<<<kB * 8, 256, 0, stream>>>(qb, kb, vb, biasb, aob);
  gemm1(aob, kHid, P(14), P(13), hgb, nullptr, hg2, kN, kHid, 1.f, 0);
  rowln_kernel<<<ceil_div(kN, 8), 256, 0, stream>>>(hg2, tln, kN, kHid);
  gemm1(tln, kHid, P(8), P(7), nullptr, nullptr, ffn, kN, 1024, 1.f, 4);
  gemm1(ffn, 1024, P(10), P(9), hg2, xcomb, xcomb, kN, kHid, 1.f, 0); // x_comb += hg

  // ---- readouts ----
  gemm(xcomb, nullptr, kHid, kHid, h0, nullptr, kHid, kHid,
       nullptr, nullptr, 0, 0, nullptr, nullptr, 0, 0,
       P(96), P(95), nullptr, nullptr, hnp, kN, kDe, 1.f, 1);          // nrp
  gemm1(hnp, kDe, P(103), P(102), nullptr, nullptr, qx, kN, kDe, 1.f, 0);
  gemm1(eacomb, kDe, P(99), P(98), nullptr, nullptr, kx, kE, kDe, 1.f, 0);
  gemm1(eacomb, kDe, P(105), P(104), nullptr, nullptr, vx, kE, kDe, 1.f, 0);
  hipMemsetAsync(m2, 0, (size_t)kN * 4 * 4, stream);
  hipMemsetAsync(z2, 0, (size_t)kN * 4 * 4, stream);
  hipMemsetAsync(agg2, 0, (size_t)kN * kDe * 4, stream);
  xscore_kernel<<<ceil_div(kE * 4, 256), 256, 0, stream>>>(qx, kx, src, s2, m2);
  xexp_kernel<<<ceil_div(kE * 4, 256), 256, 0, stream>>>(s2, m2, z2, src);
  xagg_kernel<<<ceil_div(kE * kDe, 256), 256, 0, stream>>>(s2, z2, vx, src, agg2);
  gemm1(agg2, kDe, P(101), P(100), nullptr, nullptr, smart, kN, kDe, 1.f, 0);
  pool_nodes_kernel<<<ceil_div(kB * kHid, 256), 256, 0, stream>>>(xcomb, gin);
  gemm1(gin, 512, P(22), P(21), nullptr, nullptr, gpool, kB, kDe, 1.f, 1);
  pool_mean_kernel<<<ceil_div(kB * kDe, 256), 256, 0, stream>>>(eacomb, emean, kEp, kDe, kB);
  gemm1(emean, kDe, P(5), P(4), nullptr, nullptr, gedge, kB, kDe, 1.f, 1);
  pool_mean_kernel<<<ceil_div(kB * kDe, 256), 256, 0, stream>>>(hnp, npool, kNp, kDe, kB);
  pool_mean_kernel<<<ceil_div(kB * kDe, 256), 256, 0, stream>>>(smart, spool, kNp, kDe, kB);
  gemm(npool, nullptr, kDe, kDe, spool, nullptr, kDe, kDe,
       gpool, nullptr, kDe, kDe, gedge, nullptr, kDe, kDe,
       P(24), P(23), nullptr, nullptr, h1, kB, 256, 1.f, 1);           // head1
  gemm1(h1, 256, P(26), P(25), nullptr, nullptr, h2b, kB, kDe, 1.f, 1); // head2
  head3_kernel<<<1, 128, 0, stream>>>(h2b, P(28), P(27), (float*)d_out);
#undef P
}